// FeatureTransformerSlice_33492154974540
// MI455X (gfx1250) — compile-verified
//
#include <hip/hip_runtime.h>

typedef __attribute__((ext_vector_type(2))) float v2f;
typedef __attribute__((ext_vector_type(8))) float v8f;

#define FT_BATCH       16384
#define FT_MAX_ACTIVE  32
#define FT_D           512
#define FT_BTILE       16          // batches per block
#define FT_KSLOTS      (FT_BTILE * FT_MAX_ACTIVE)   // 512 virtual K slots
#define FT_DPERBLOCK   128         // 8 waves * 16 columns

// out[b, d] = bias[d] + sum_k val[b,k] * W[idx[b,k], d]
// Computed as block-diagonal A[16 x 512] (values) times gathered B[512 x 16]
// (weight row segments) via V_WMMA_F32_16X16X4_F32, f32 in / f32 accumulate.
__global__ __launch_bounds__(256) void ft_embedbag_wmma_kernel(
    const int*   __restrict__ fidx,    // [BATCH, MAX_ACTIVE]
    const float* __restrict__ fval,    // [BATCH, MAX_ACTIVE]
    const float* __restrict__ weight,  // [NUM_INPUTS, 512]
    const float* __restrict__ bias,    // [512]
    float*       __restrict__ out)     // [BATCH, 512]
{
    __shared__ int   s_idx[FT_KSLOTS];
    __shared__ float s_val[FT_KSLOTS];

    const int btile = blockIdx.x;            // 0..1023
    const int dblk  = blockIdx.y;            // 0..3
    const int tid   = threadIdx.x;

    // Stage this batch-tile's indices/values (flat: local_b*32 + k).
    {
        const int base = btile * FT_KSLOTS;
        #pragma unroll
        for (int i = 0; i < FT_KSLOTS; i += 256) {
            s_idx[i + tid] = fidx[base + i + tid];
            s_val[i + tid] = fval[base + i + tid];
        }
    }
    __syncthreads();

    const int wave = tid >> 5;
    const int lane = tid & 31;
    const int n    = lane & 15;              // N column within 16x16 tile (also M for A)
    const int half = lane >> 4;              // lane group selects K pair {0,1} vs {2,3}
    const int d0   = dblk * FT_DPERBLOCK + wave * 16;
    const unsigned dn = (unsigned)(d0 + n);  // this lane's weight/output column

    // Accumulator seeded with bias[d] (bias depends only on N -> same in all 8 VGPRs).
    const float bv = bias[dn];
    v8f c;
    #pragma unroll
    for (int j = 0; j < 8; ++j) c[j] = bv;

    // 128 chunks of K=4 walk the 512 (batch,k) slots. Chunk ch covers
    // global K = 4*ch..4*ch+3, owned by local batch bown = ch/8.
    // Weight offsets fit in 32 bits (84 MB) -> GVS saddr+voffset addressing.
    #pragma unroll 4
    for (int ch = 0; ch < 128; ++ch) {
        const int bown  = ch >> 3;
        const int kk    = ((ch & 7) << 2) + (half << 1);  // within-batch k for this lane's K pair
        const int sbase = (bown << 5) + kk;

        const float av0 = s_val[sbase];
        const float av1 = s_val[sbase + 1];
        const unsigned r0 = (unsigned)s_idx[sbase];
        const unsigned r1 = (unsigned)s_idx[sbase + 1];

        // A fragment: block-diagonal values. Lane holds A[m = n, K = kk(+1)].
        v2f a;
        a.x = (n == bown) ? av0 : 0.0f;
        a.y = (n == bown) ? av1 : 0.0f;

        // B fragment: gathered weight row segments. Lane holds B[K = kk(+1), n].
        v2f b;
        b.x = weight[r0 * (unsigned)FT_D + dn];
        b.y = weight[r1 * (unsigned)FT_D + dn];

        c = __builtin_amdgcn_wmma_f32_16x16x4_f32(
                /*neg_a=*/false, a, /*neg_b=*/false, b,
                /*c_mod=*/(short)0, c, /*reuse_a=*/false, /*reuse_b=*/false);
    }

    // C/D layout: VGPR j -> row m = j + 8*half, column n.
    const unsigned orow = (unsigned)(btile * FT_BTILE + half * 8) * (unsigned)FT_D + dn;
    #pragma unroll
    for (int j = 0; j < 8; ++j)
        out[orow + (unsigned)j * (unsigned)FT_D] = c[j];
}

extern "C" void kernel_launch(void* const* d_in, const int* in_sizes, int n_in,
                              void* d_out, int out_size, void* d_ws, size_t ws_size,
                              hipStream_t stream) {
    (void)in_sizes; (void)n_in; (void)out_size; (void)d_ws; (void)ws_size;
    const int*   fidx   = (const int*)  d_in[0];  // feature_indices [16384,32] int32
    const float* fval   = (const float*)d_in[1];  // feature_values  [16384,32] f32
    const float* weight = (const float*)d_in[2];  // weight [41024,512] f32
    const float* bias   = (const float*)d_in[3];  // bias [512] f32
    float*       out    = (float*)d_out;          // [16384,512] f32

    dim3 grid(FT_BATCH / FT_BTILE, FT_D / FT_DPERBLOCK);  // 1024 x 4
    ft_embedbag_wmma_kernel<<<grid, 256, 0, stream>>>(fidx, fval, weight, bias, out);
}